// TimeShiftedMultiModalAttention_86629490360328
// MI455X (gfx1250) — compile-verified
//
#include <hip/hip_runtime.h>

// ---- problem constants ----
#define T_SEQ   2048
#define D_MODEL 512
#define N_HEADS 8
#define DIM_H   64
#define BH      32        // B * heads
#define M_ROWS  8192      // B * T
#define MAX_LAG 3
#define SCALE   0.125f    // 64^-0.5

typedef __attribute__((ext_vector_type(2))) float        v2f;
typedef __attribute__((ext_vector_type(4))) float        v4f;
typedef __attribute__((ext_vector_type(8))) float        v8f;
typedef __attribute__((ext_vector_type(4))) int          v4i;
typedef __attribute__((ext_vector_type(8))) int          v8i;
typedef __attribute__((ext_vector_type(4))) unsigned int v4u;

typedef __attribute__((address_space(1))) v4i* gv4i_p;   // global
typedef __attribute__((address_space(3))) v4i* lv4i_p;   // LDS

// ---------------------------------------------------------------------
// CDNA5 async global->LDS copy (ASYNCcnt-tracked).
// ---------------------------------------------------------------------
#if defined(__has_builtin) && __has_builtin(__builtin_amdgcn_global_load_async_to_lds_b128)
__device__ __forceinline__ void async_cp16(const float* g, float* l) {
    float* gm = const_cast<float*>(g);
    __builtin_amdgcn_global_load_async_to_lds_b128(
        (gv4i_p)gm, (lv4i_p)l, 0, 0);
}
#else
__device__ __forceinline__ void async_cp16(const float* g, float* l) {
    unsigned lds_off =
        (unsigned)(__SIZE_TYPE__)(__attribute__((address_space(3))) float*)l;
    unsigned long long ga = (unsigned long long)g;
    asm volatile("global_load_async_to_lds_b128 %0, %1, off"
                 :: "v"(lds_off), "v"(ga) : "memory");
}
#endif

#if defined(__has_builtin) && __has_builtin(__builtin_amdgcn_s_wait_asynccnt)
#define WAIT_ASYNC(n) __builtin_amdgcn_s_wait_asynccnt(n)
#else
#define WAIT_ASYNC(n) asm volatile("s_wait_asynccnt %0" :: "n"(n) : "memory")
#endif

#if defined(__has_builtin) && __has_builtin(__builtin_amdgcn_s_wait_tensorcnt)
#define WAIT_TENSOR(n) __builtin_amdgcn_s_wait_tensorcnt(n)
#else
#define WAIT_TENSOR(n) asm volatile("s_wait_tensorcnt %0" :: "n"(n) : "memory")
#endif

// ---------------------------------------------------------------------
// Tensor Data Mover: load a 64x64 fp32 tile (row stride 512 floats) from
// global memory into LDS with ONE instruction (TENSORcnt-tracked).
// D# layout per CDNA5 ISA ch.8 (group0 128b, group1 256b, groups2/3 = 0
// for a 2-D tensor). This toolchain uses the 6-arg builtin form.
// ---------------------------------------------------------------------
#if defined(__has_builtin) && __has_builtin(__builtin_amdgcn_tensor_load_to_lds)
#define HAVE_TDM 1
__device__ __forceinline__ void tdm_load_tile_64x64(const float* gsrc, float* ldst) {
    unsigned long long ga = (unsigned long long)gsrc;
    unsigned lds =
        (unsigned)(__SIZE_TYPE__)(__attribute__((address_space(3))) float*)ldst;
    v4u g0;
    g0.x = 1u;                                   // count=1, user mode, no gather
    g0.y = lds;                                  // lds_addr  (bits 63:32)
    g0.z = (unsigned)ga;                         // global_addr[31:0]  (bits 95:64)
    g0.w = (unsigned)((ga >> 32) & 0x01ffffffu)  // global_addr[56:32] (bits 120:96)
         | (2u << 30);                           // type=2 ("image")
    v8i g1;
    g1[0] = 0x00020000;          // workgroup_mask=0, data_size=2 (4 bytes)
    g1[1] = (int)(512u << 16);   // tensor_dim0[15:0]=512  (bits 63:48)
    g1[2] = (int)(512u << 16);   // tensor_dim0 hi=0 | tensor_dim1[15:0]=512
    g1[3] = (int)(64u << 16);    // tensor_dim1 hi=0 | tile_dim0=64
    g1[4] = 64;                  // tile_dim1=64, tile_dim2=0
    g1[5] = 512;                 // tensor_dim0_stride[31:0]=512
    g1[6] = 0;                   // stride0 hi, stride1 lo
    g1[7] = 0;                   // stride1 hi
    v4i z4; z4.x = 0; z4.y = 0; z4.z = 0; z4.w = 0;   // groups 2/3: 2-D tensor
    v8i z8;
    #pragma unroll
    for (int i = 0; i < 8; ++i) z8[i] = 0;
    __builtin_amdgcn_tensor_load_to_lds(g0, g1, z4, z4, z8, 0);
}
#else
#define HAVE_TDM 0
#endif

// =====================================================================
// GEMM 1: C = A[8192,512] @ W[512,512]  -> stored split-head layout
//         dst[((b*8+head)*2048 + t)*64 + dim]
// block = 128 threads (4 waves). Block tile 64(M) x 64(N), wave = 16x64.
// W staged in double-buffered 64x64 LDS tiles via the TENSOR DATA MOVER:
// wave 0 issues one tensor_load_to_lds per K-chunk (TENSORcnt-pipelined).
// =====================================================================
__global__ __launch_bounds__(128)
void gemm_qkv_kernel(const float* __restrict__ A, const float* __restrict__ W,
                     float* __restrict__ dst)
{
    __shared__ float ldsW[2][64 * 64];     // 2 x 16 KB

    const int tid  = threadIdx.x;
    const int lane = tid & 31;
    const int wave = tid >> 5;
    const int half = lane >> 4;            // K-half select for f32 WMMA frags
    const int l16  = lane & 15;

    const int m0 = blockIdx.x * 64 + wave * 16;
    const int n0 = blockIdx.y * 64;
    const int arow = m0 + l16;

    v8f acc[4] = {};

#if HAVE_TDM
    if (wave == 0) tdm_load_tile_64x64(&W[n0], ldsW[0]);
#else
    #pragma unroll
    for (int p = 0; p < 8; ++p) {
        int f = tid + 128 * p, kk = f >> 4, nq = f & 15;
        async_cp16(&W[kk * D_MODEL + n0 + nq * 4], &ldsW[0][kk * 64 + nq * 4]);
    }
#endif

    for (int it = 0; it < 8; ++it) {
        const int kc = it * 64;
#if HAVE_TDM
        if (wave == 0) {
            if (it + 1 < 8) {
                tdm_load_tile_64x64(&W[(kc + 64) * D_MODEL + n0],
                                    ldsW[(it + 1) & 1]);
                WAIT_TENSOR(1);            // drain current tile, next in flight
            } else {
                WAIT_TENSOR(0);
            }
        }
        __builtin_prefetch(&A[arow * D_MODEL + kc + 64], 0, 0);
#else
        if (it + 1 < 8) {
            float* nb_buf = ldsW[(it + 1) & 1];
            #pragma unroll
            for (int p = 0; p < 8; ++p) {
                int f = tid + 128 * p, kk = f >> 4, nq = f & 15;
                async_cp16(&W[(kc + 64 + kk) * D_MODEL + n0 + nq * 4],
                           &nb_buf[kk * 64 + nq * 4]);
            }
            WAIT_ASYNC(8);
        } else {
            WAIT_ASYNC(0);
        }
#endif
        __syncthreads();

        const float* wb = ldsW[it & 1];
        #pragma unroll 4
        for (int ki = 0; ki < 64; ki += 4) {
            const int klane = ki + half * 2;
            v2f a = *(const v2f*)&A[arow * D_MODEL + kc + klane];
            #pragma unroll
            for (int nb = 0; nb < 4; ++nb) {
                const int col = nb * 16 + l16;
                v2f b;
                b.x = wb[klane * 64 + col];
                b.y = wb[(klane + 1) * 64 + col];
                acc[nb] = __builtin_amdgcn_wmma_f32_16x16x4_f32(
                    false, a, false, b, (short)0, acc[nb], false, false);
            }
        }
        __syncthreads();
    }

    // store: split-head scatter. C layout: VGPR r -> M = m0 + r + half*8
    #pragma unroll
    for (int nb = 0; nb < 4; ++nb) {
        const int col  = n0 + nb * 16 + l16;
        const int head = col >> 6;
        const int dim  = col & 63;
        #pragma unroll
        for (int r = 0; r < 8; ++r) {
            const int row = m0 + r + half * 8;
            const int bb  = row >> 11;
            const int t   = row & 2047;
            dst[(((size_t)(bb * N_HEADS + head) * T_SEQ) + t) * DIM_H + dim] =
                acc[nb][r];
        }
    }
}

// =====================================================================
// GEMM 2: out = headmerge(H)[8192,512] @ Wout[512,512] + bout
// A gathered from split-head layout; C stored contiguous [B,T,D].
// W staged via double-buffered ASYNC global->LDS copies (ASYNCcnt).
// =====================================================================
__global__ __launch_bounds__(128)
void gemm_out_kernel(const float* __restrict__ H, const float* __restrict__ Wout,
                     const float* __restrict__ bout, float* __restrict__ out)
{
    __shared__ float ldsW[2][64 * 64];

    const int tid  = threadIdx.x;
    const int lane = tid & 31;
    const int wave = tid >> 5;
    const int half = lane >> 4;
    const int l16  = lane & 15;

    const int m0 = blockIdx.x * 64 + wave * 16;
    const int n0 = blockIdx.y * 64;
    const int arow = m0 + l16;
    const int ab   = arow >> 11;
    const int at   = arow & 2047;

    float bias[4];
    #pragma unroll
    for (int nb = 0; nb < 4; ++nb) bias[nb] = bout[n0 + nb * 16 + l16];

    v8f acc[4] = {};

    #pragma unroll
    for (int p = 0; p < 8; ++p) {
        int f = tid + 128 * p, kk = f >> 4, nq = f & 15;
        async_cp16(&Wout[kk * D_MODEL + n0 + nq * 4], &ldsW[0][kk * 64 + nq * 4]);
    }

    for (int it = 0; it < 8; ++it) {
        const int kc = it * 64;
        if (it + 1 < 8) {
            float* nb_buf = ldsW[(it + 1) & 1];
            #pragma unroll
            for (int p = 0; p < 8; ++p) {
                int f = tid + 128 * p, kk = f >> 4, nq = f & 15;
                async_cp16(&Wout[(kc + 64 + kk) * D_MODEL + n0 + nq * 4],
                           &nb_buf[kk * 64 + nq * 4]);
            }
            WAIT_ASYNC(8);
        } else {
            WAIT_ASYNC(0);
        }
        __syncthreads();

        const float* wb = ldsW[it & 1];
        #pragma unroll 4
        for (int ki = 0; ki < 64; ki += 4) {
            const int klane = ki + half * 2;
            const int kgl   = kc + klane;          // even -> pair stays in head
            const int head  = kgl >> 6;
            const int dim   = kgl & 63;
            v2f a = *(const v2f*)&H[(((size_t)(ab * N_HEADS + head) * T_SEQ) + at)
                                        * DIM_H + dim];
            #pragma unroll
            for (int nb = 0; nb < 4; ++nb) {
                const int col = nb * 16 + l16;
                v2f b;
                b.x = wb[klane * 64 + col];
                b.y = wb[(klane + 1) * 64 + col];
                acc[nb] = __builtin_amdgcn_wmma_f32_16x16x4_f32(
                    false, a, false, b, (short)0, acc[nb], false, false);
            }
        }
        __syncthreads();
    }

    #pragma unroll
    for (int nb = 0; nb < 4; ++nb) {
        const int col = n0 + nb * 16 + l16;
        #pragma unroll
        for (int r = 0; r < 8; ++r) {
            const int row = m0 + r + half * 8;
            out[(size_t)row * D_MODEL + col] = acc[nb][r] + bias[nb];
        }
    }
}

// =====================================================================
// Band attention: window of <=4 keys per query.
// One wave per (bh, i); lane owns dims [2*lane, 2*lane+1] of the 64-dim head.
// =====================================================================
__global__ __launch_bounds__(256)
void band_attn_kernel(const float* __restrict__ q, const float* __restrict__ k,
                      const float* __restrict__ v, const float* __restrict__ lagw,
                      float* __restrict__ attn, float* __restrict__ hout)
{
    const int lane = threadIdx.x;                    // 0..31
    const int p    = blockIdx.x * 8 + threadIdx.y;   // 0..65535
    const int bh   = p >> 11;
    const int i    = p & 2047;

    const size_t rowq = ((size_t)bh * T_SEQ + i) * DIM_H;
    v2f qv = *(const v2f*)&q[rowq + lane * 2];

    float sc[MAX_LAG + 1];
    #pragma unroll
    for (int l = 0; l <= MAX_LAG; ++l) {
        const int j = i - l;
        float s = -3.0e38f;
        if (j >= 0) {
            v2f kv = *(const v2f*)&k[((size_t)bh * T_SEQ + j) * DIM_H + lane * 2];
            float part = qv.x * kv.x + qv.y * kv.y;
            #pragma unroll
            for (int off = 16; off >= 1; off >>= 1)
                part += __shfl_xor(part, off, 32);
            s = part * SCALE + lagw[l];
        }
        sc[l] = s;
    }

    float mx = fmaxf(fmaxf(sc[0], sc[1]), fmaxf(sc[2], sc[3]));
    float e[MAX_LAG + 1];
    float sum = 0.0f;
    #pragma unroll
    for (int l = 0; l <= MAX_LAG; ++l) { e[l] = __expf(sc[l] - mx); sum += e[l]; }
    const float inv = 1.0f / sum;

    v2f o; o.x = 0.0f; o.y = 0.0f;
    #pragma unroll
    for (int l = 0; l <= MAX_LAG; ++l) {
        const int j = i - l;
        if (j >= 0) {
            v2f vv = *(const v2f*)&v[((size_t)bh * T_SEQ + j) * DIM_H + lane * 2];
            const float pr = e[l] * inv;
            o.x += pr * vv.x;
            o.y += pr * vv.y;
        }
    }
    *(v2f*)&hout[rowq + lane * 2] = o;

    if (lane <= MAX_LAG && (i - lane) >= 0) {
        attn[(size_t)bh * T_SEQ * T_SEQ + (size_t)i * T_SEQ + (i - lane)] =
            e[lane] * inv;
    }
}

// =====================================================================
// Zero-fill the 537 MB attn output region (banned entries are exact 0.0).
// =====================================================================
__global__ __launch_bounds__(256)
void zero_attn_kernel(v4f* __restrict__ p, long n4)
{
    v4f z; z.x = 0.0f; z.y = 0.0f; z.z = 0.0f; z.w = 0.0f;
    for (long idx = (long)blockIdx.x * blockDim.x + threadIdx.x; idx < n4;
         idx += (long)gridDim.x * blockDim.x)
        p[idx] = z;
}

extern "C" void kernel_launch(void* const* d_in, const int* in_sizes, int n_in,
                              void* d_out, int out_size, void* d_ws, size_t ws_size,
                              hipStream_t stream) {
    const float* x    = (const float*)d_in[0];
    const float* Wq   = (const float*)d_in[1];
    const float* Wk   = (const float*)d_in[2];
    const float* Wv   = (const float*)d_in[3];
    const float* Wout = (const float*)d_in[4];
    const float* bout = (const float*)d_in[5];
    const float* lagw = (const float*)d_in[6];

    float* out  = (float*)d_out;                              // [4,2048,512]
    float* attn = out + (size_t)4 * T_SEQ * D_MODEL;          // [32,2048,2048]

    const size_t per = (size_t)BH * T_SEQ * DIM_H;            // 4.19M floats
    float* qw = (float*)d_ws;
    float* kw = qw + per;
    float* vw = kw + per;
    float* ow = vw + per;

    dim3 gemm_grid(M_ROWS / 64, D_MODEL / 64);                // (128, 8)
    gemm_qkv_kernel<<<gemm_grid, 128, 0, stream>>>(x, Wq, qw);
    gemm_qkv_kernel<<<gemm_grid, 128, 0, stream>>>(x, Wk, kw);
    gemm_qkv_kernel<<<gemm_grid, 128, 0, stream>>>(x, Wv, vw);

    const long n4 = ((long)BH * T_SEQ * T_SEQ) / 4;           // 33,554,432 float4
    zero_attn_kernel<<<4096, 256, 0, stream>>>((v4f*)attn, n4);

    dim3 band_blk(32, 8);
    band_attn_kernel<<<(BH * T_SEQ) / 8, band_blk, 0, stream>>>(
        qw, kw, vw, lagw, attn, ow);

    gemm_out_kernel<<<gemm_grid, 128, 0, stream>>>(ow, Wout, bout, out);
}